// SSDMatcher_11467562680933
// MI455X (gfx1250) — compile-verified
//
#include <hip/hip_runtime.h>
#include <hip/hip_bf16.h>
#include <stdint.h>

// Problem constants from the reference setup_inputs()
#define BATCH 8
#define NANCH 32768
#define NGT   100

// Monotonic (totally ordered) unsigned mapping of IEEE float:
// key(a) < key(b)  <=>  a < b  (for all finite values; GIoU in [-1,1])
__device__ __forceinline__ unsigned float_orderable(float v) {
    unsigned u = __float_as_uint(v);
    return (u & 0x80000000u) ? ~u : (u | 0x80000000u);
}

__global__ void ssd_ws_init(unsigned long long* ws) {
    int i = blockIdx.x * blockDim.x + threadIdx.x;
    if (i < BATCH * NGT) ws[i] = 0ull;
}

// grid = (NANCH/256, BATCH), block = 256 threads (8 wave32 per block)
__global__ void ssd_match_kernel(const float* __restrict__ pred_boxes,  // [B,A,4] cxcywh
                                 const float* __restrict__ gt_boxes,    // [B,G,4] cxcywh
                                 float* __restrict__ out_val,           // [B,A]
                                 int*   __restrict__ out_idx,           // [B,A]
                                 unsigned long long* __restrict__ ws)   // [B,G] packed max
{
    const int tid = threadIdx.x;
    const int b   = blockIdx.y;
    const int a   = blockIdx.x * blockDim.x + tid;

    // Prefetch this thread's pred box while we stage GT boxes into LDS.
    const float* pb_ptr = pred_boxes + (((size_t)b * NANCH + a) << 2);
    __builtin_prefetch(pb_ptr, 0, 0);   // -> global_prefetch_b8

    // ---- Stage GT boxes (100 x float4 = 1600B) into LDS with async DMA ----
    __shared__ float4 s_raw[NGT];
    __shared__ float  s_x1[NGT], s_y1[NGT], s_x2[NGT], s_y2[NGT], s_area[NGT];

    if (tid < NGT) {
        unsigned           lds_off = (unsigned)(uintptr_t)(&s_raw[tid]);
        unsigned long long gaddr   = (unsigned long long)(uintptr_t)
                                     (gt_boxes + (((size_t)b * NGT + tid) << 2));
        // CDNA5 async global->LDS copy (ASYNCcnt-tracked), GV addressing mode.
        asm volatile("global_load_async_to_lds_b128 %0, %1, off"
                     :: "v"(lds_off), "v"(gaddr) : "memory");
    }
    asm volatile("s_wait_asynccnt 0" ::: "memory");
    __syncthreads();

    // Convert staged cxcywh -> corner SoA + area (broadcast-friendly layout).
    if (tid < NGT) {
        float4 r = s_raw[tid];
        float x1 = r.x - 0.5f * r.z, y1 = r.y - 0.5f * r.w;
        float x2 = r.x + 0.5f * r.z, y2 = r.y + 0.5f * r.w;
        s_x1[tid] = x1; s_y1[tid] = y1; s_x2[tid] = x2; s_y2[tid] = y2;
        s_area[tid] = (x2 - x1) * (y2 - y1);
    }
    __syncthreads();

    // This thread's pred box, corner form.
    const float4 p = *reinterpret_cast<const float4*>(pb_ptr);   // global_load_b128
    const float px1 = p.x - 0.5f * p.z, py1 = p.y - 0.5f * p.w;
    const float px2 = p.x + 0.5f * p.z, py2 = p.y + 0.5f * p.w;
    const float parea = (px2 - px1) * (py2 - py1);

    float best_v   = -3.0f;   // GIoU >= -1 always
    int   best_g   = 0;
    const int lane = tid & (warpSize - 1);

    for (int g = 0; g < NGT; ++g) {
        // LDS broadcast reads: every lane reads the same address (no conflicts).
        const float gx1 = s_x1[g], gy1 = s_y1[g], gx2 = s_x2[g], gy2 = s_y2[g];
        const float garea = s_area[g];

        const float iw = fminf(px2, gx2) - fmaxf(px1, gx1);
        const float ih = fminf(py2, gy2) - fmaxf(py1, gy1);
        const float inter = fmaxf(iw, 0.0f) * fmaxf(ih, 0.0f);
        const float uni   = parea + garea - inter;

        const float ew = fmaxf(px2, gx2) - fminf(px1, gx1);
        const float eh = fmaxf(py2, gy2) - fminf(py1, gy1);
        const float enc = fmaxf(ew, 0.0f) * fmaxf(eh, 0.0f);

        // Two fast reciprocals (v_rcp_f32, TRANS pipe co-executes with VALU).
        const float giou = inter * __builtin_amdgcn_rcpf(uni)
                         - (enc - uni) * __builtin_amdgcn_rcpf(enc);

        // per-anchor argmax over g: strictly-greater keeps the first max (jnp.argmax)
        if (giou > best_v) { best_v = giou; best_g = g; }

        // per-GT argmax over anchors: pack (orderable(val), ~anchor) so that ties
        // pick the LOWEST anchor index; wave32 shuffle-reduce then one u64 atomic.
        unsigned long long key =
            ((unsigned long long)float_orderable(giou) << 32) | (unsigned)(~a);
        #pragma unroll
        for (int off = 16; off > 0; off >>= 1) {
            unsigned long long o = __shfl_xor(key, off, 32);
            key = (o > key) ? o : key;
        }
        if (lane == 0)
            atomicMax(&ws[b * NGT + g], key);   // global_atomic_max_u64
    }

    const size_t oi = (size_t)b * NANCH + a;
    out_val[oi] = best_v;
    out_idx[oi] = best_g;
}

// Deterministic scatter: each GT (in ascending g) claims its best anchor.
// One thread per batch -> sequential last-write-wins like .at[idx].set order.
__global__ void ssd_scatter_kernel(const unsigned long long* __restrict__ ws,
                                   float* __restrict__ out_val,
                                   int*   __restrict__ out_idx)
{
    int b = threadIdx.x;
    if (b >= BATCH) return;
    for (int g = 0; g < NGT; ++g) {
        unsigned long long key = ws[b * NGT + g];
        unsigned a = ~(unsigned)(key & 0xFFFFFFFFull);
        out_idx[(size_t)b * NANCH + a] = g;
        out_val[(size_t)b * NANCH + a] = 1.0f;
    }
}

extern "C" void kernel_launch(void* const* d_in, const int* in_sizes, int n_in,
                              void* d_out, int out_size, void* d_ws, size_t ws_size,
                              hipStream_t stream) {
    const float* pred_boxes = (const float*)d_in[0];
    // d_in[1] = pred_logits: dead in the reference (sigmoid result unused) -> skipped
    const float* gt_boxes   = (const float*)d_in[2];
    // d_in[3] = gt_labels: unused by the matcher outputs

    float* out_val = (float*)d_out;                       // [B,A] float32
    int*   out_idx = (int*)((float*)d_out + (size_t)BATCH * NANCH);  // [B,A] int32 bits

    unsigned long long* ws = (unsigned long long*)d_ws;   // [B,G] packed (val,~idx)

    ssd_ws_init<<<(BATCH * NGT + 255) / 256, 256, 0, stream>>>(ws);

    dim3 grid(NANCH / 256, BATCH);
    ssd_match_kernel<<<grid, 256, 0, stream>>>(pred_boxes, gt_boxes,
                                               out_val, out_idx, ws);

    ssd_scatter_kernel<<<1, 32, 0, stream>>>(ws, out_val, out_idx);
}